// L2Conv3D_53300544144072
// MI455X (gfx1250) — compile-verified
//
#include <hip/hip_runtime.h>
#include <hip/hip_bf16.h>
#include <math.h>

typedef __attribute__((ext_vector_type(2))) float v2f;
typedef __attribute__((ext_vector_type(8))) float v8f;

#define BATCH   16
#define CIN     128
#define DIN     24
#define DIN3    (DIN * DIN * DIN)
#define NF      128
#define KVOL    27
#define KTOT    (CIN * KVOL)          // 3456 = 54 * 64
#define DOUT    22
#define SOUT    (DOUT * DOUT * DOUT)  // 10648
#define KCHUNK  64
#define MTILE   64                    // spatial positions per block
#define WS      (KCHUNK + 2)          // padded weight row stride
#define SPR4    (4 * MTILE + 4)       // patch kk-quad row stride: 260 (= 4 mod 64)

__global__ __launch_bounds__(256)
void l2conv3d_wmma_f32(const float* __restrict__ xs,
                       const float* __restrict__ fv,
                       float* __restrict__ out) {
    __shared__ float sW[NF * WS];              // weights [p][kk]          : 33792 B
    __shared__ float sP[(KCHUNK / 4) * SPR4];  // patches [kk>>2][m][kk&3] : 16640 B
    __shared__ float sProtoN[NF];
    __shared__ float sPatchN[MTILE];

    const int t    = threadIdx.x;
    const int wave = t >> 5;
    const int lane = t & 31;
    const int half = lane >> 4;        // K sub-pair select per ISA A/B layout
    const int lrow = lane & 15;
    const int waveM = wave & 3;        // 4 proto groups of 32
    const int waveN = wave >> 2;       // 2 spatial groups of 32
    const int pbase   = waveM * 32;
    const int sbase_l = waveN * 32;

    const int b      = blockIdx.y;
    const int s_base = blockIdx.x * MTILE;

    // ---- chunk-invariant im2col precompute --------------------------------
    // staging mapping: thread owns fixed kk = t&63, and m = (t>>6) + 4*j
    const int kkf  = t & (KCHUNK - 1);
    const int mgrp = t >> 6;           // 0..3
    int moff[16];                      // patch-origin offset (z*DIN + y)*DIN + x
    unsigned mvalbits = 0u;
    #pragma unroll
    for (int j = 0; j < 16; ++j) {
        int m = mgrp + 4 * j;
        int s = s_base + m;
        int valid = (s < SOUT) ? 1 : 0;
        int sc = valid ? s : 0;        // clamp -> always in-range address
        int z = sc / (DOUT * DOUT);
        int r = sc - z * (DOUT * DOUT);
        int y = r / DOUT;
        int x = r - y * DOUT;
        moff[j] = (z * DIN + y) * DIN + x;
        mvalbits |= ((unsigned)valid) << j;
    }
    const long xbatch = (long)b * CIN * DIN3;

    v8f acc00 = {}, acc01 = {}, acc10 = {}, acc11 = {};
    float pn = 0.0f;   // ||proto||^2 (threads t<128 own p=t)
    float an = 0.0f;   // ||patch||^2 (threads t<64  own m=t)

    for (int k0 = 0; k0 < KTOT; k0 += KCHUNK) {
        __syncthreads();  // previous chunk fully consumed before overwrite

        // ---- stage weights: sW[p][kk] = fv[p*KTOT + k0 + kk] (coalesced 128B runs)
        {
            int p   = t >> 1;
            int kkb = (t & 1) * 32;
            const float* g  = &fv[p * KTOT + k0 + kkb];
            float*       sh = &sW[p * WS + kkb];
            #pragma unroll
            for (int j = 0; j < 32; ++j) sh[j] = g[j];
        }

        // ---- stage patches: one k-decomposition per chunk, 16 clean loads,
        //      K-quad interleaved layout, conflict-free stores
        {
            int k  = k0 + kkf;
            int c  = k / KVOL;
            int rk = k - c * KVOL;
            int dz = rk / 9;
            int r2 = rk - dz * 9;
            int dy = r2 / 3;
            int dx = r2 - dy * 3;
            const float* xb = xs + xbatch + (long)c * DIN3
                              + (dz * DIN + dy) * DIN + dx;
            float* sp = &sP[(kkf >> 2) * SPR4 + (kkf & 3)];
            #pragma unroll
            for (int j = 0; j < 16; ++j) {
                float v = xb[moff[j]];                       // always in-range
                if (!((mvalbits >> j) & 1u)) v = 0.0f;       // mask tail positions
                sp[(mgrp + 4 * j) * 4] = v;
            }
        }

        // prefetch next weight chunk (gfx1250 global_prefetch_b8 path)
        if (k0 + KCHUNK < KTOT)
            __builtin_prefetch(&fv[(t >> 1) * KTOT + k0 + KCHUNK + (t & 1) * 32], 0, 1);

        __syncthreads();

        // ---- norms from staged tiles
        if (t < NF) {
            #pragma unroll 8
            for (int kk = 0; kk < KCHUNK; ++kk) {
                float v = sW[t * WS + kk];
                pn = fmaf(v, v, pn);
            }
        }
        if (t < MTILE) {
            #pragma unroll 4
            for (int kq = 0; kq < KCHUNK / 4; ++kq) {
                const float* pr = &sP[kq * SPR4 + t * 4];
                #pragma unroll
                for (int i = 0; i < 4; ++i) {
                    float v = pr[i];
                    an = fmaf(v, v, an);
                }
            }
        }

        // ---- 2x2 register-tiled WMMA sweep: 64 wmma / chunk / wave
        #pragma unroll
        for (int kc = 0; kc < KCHUNK; kc += 4) {
            // A 16x4 f32 (protos): pair-contiguous along kk -> v2f loads
            const int ka = kc + 2 * half;
            v2f a0 = *(const v2f*)&sW[(pbase + lrow) * WS + ka];
            v2f a1 = *(const v2f*)&sW[(pbase + 16 + lrow) * WS + ka];
            // B 4x16 f32 (patches): quad-interleaved -> adjacent pair per lane,
            // b0/b1 256B apart -> single ds_load_2addr_b64, zero repack
            const float* pb = &sP[(kc >> 2) * SPR4];
            const int bo = (sbase_l + lrow) * 4 + 2 * half;
            v2f b0 = *(const v2f*)&pb[bo];
            v2f b1 = *(const v2f*)&pb[bo + 64];

            acc00 = __builtin_amdgcn_wmma_f32_16x16x4_f32(false, a0, false, b0, (short)0, acc00, false, false);
            acc01 = __builtin_amdgcn_wmma_f32_16x16x4_f32(false, a0, false, b1, (short)0, acc01, false, false);
            acc10 = __builtin_amdgcn_wmma_f32_16x16x4_f32(false, a1, false, b0, (short)0, acc10, false, false);
            acc11 = __builtin_amdgcn_wmma_f32_16x16x4_f32(false, a1, false, b1, (short)0, acc11, false, false);
        }
    }

    if (t < NF)    sProtoN[t] = pn;
    if (t < MTILE) sPatchN[t] = an;
    __syncthreads();

    // ---- combine + store: rows = protos (M = v + 8*half), cols = spatial (N = lrow)
    // consecutive lanes write consecutive s -> coalesced 64B runs
    auto emit = [&](const v8f& a, int pofs, int sofs) {
        #pragma unroll
        for (int v = 0; v < 8; ++v) {
            int p  = pbase + pofs + v + 8 * half;
            int sl = sbase_l + sofs + lrow;
            int s  = s_base + sl;
            if (s < SOUT) {
                float d = sPatchN[sl] + sProtoN[p] - 2.0f * a[v];
                out[((long)(b * NF + p)) * SOUT + s] = sqrtf(fabsf(d) + 1e-14f);
            }
        }
    };
    emit(acc00, 0, 0);
    emit(acc01, 0, 16);
    emit(acc10, 16, 0);
    emit(acc11, 16, 16);
}

extern "C" void kernel_launch(void* const* d_in, const int* in_sizes, int n_in,
                              void* d_out, int out_size, void* d_ws, size_t ws_size,
                              hipStream_t stream) {
    const float* xs = (const float*)d_in[0];        // [16,128,24,24,24] f32
    const float* fv = (const float*)d_in[1];        // [128,128,3,3,3]  f32
    float* out = (float*)d_out;                     // [16,128,22,22,22] f32

    dim3 grid((SOUT + MTILE - 1) / MTILE, BATCH);   // 167 x 16 blocks
    dim3 block(256);                                // 8 waves (wave32)
    l2conv3d_wmma_f32<<<grid, block, 0, stream>>>(xs, fv, out);
}